// FALR3_35794257445091
// MI455X (gfx1250) — compile-verified
//
#include <hip/hip_runtime.h>
#include <float.h>

#define B_ 2
#define N_ 512
#define H_ 128
#define D_ 128
#define T_ 8

#define ICHUNK 64
#define NWAVE 9          // waves 0..7: msg_e d-tiles; wave 8: triplet (te) tile
#define BDIM (NWAVE * 32)
#define EPITCH 132       // 128 K + 4 pad: 16B-aligned rows, bank-conflict-free A reads

typedef float v2f __attribute__((ext_vector_type(2)));
typedef float v8f __attribute__((ext_vector_type(8)));

// sign-aware float atomic max via integer atomics (correct IEEE ordering)
__device__ __forceinline__ void atomicMaxF32(float* addr, float v) {
    if (v >= 0.0f) atomicMax((int*)addr, __float_as_int(v));
    else           atomicMin((unsigned int*)addr, __float_as_uint(v));
}

// ---- CDNA5 async global->LDS copies (ASYNCcnt-tracked, no VGPR round trip) ----
__device__ __forceinline__ void async_ld_b128x2(unsigned lds_off, const void* gaddr) {
    // copies 32 contiguous bytes: two B128 transfers sharing addr regs via inst offset
    asm volatile("global_load_async_to_lds_b128 %0, %1, off\n\t"
                 "global_load_async_to_lds_b128 %0, %1, off offset:16"
                 :: "v"(lds_off), "v"((unsigned long long)gaddr) : "memory");
}
__device__ __forceinline__ void async_ld_b32(unsigned lds_off, const void* gaddr) {
    asm volatile("global_load_async_to_lds_b32 %0, %1, off"
                 :: "v"(lds_off), "v"((unsigned long long)gaddr) : "memory");
}
__device__ __forceinline__ void wait_async0() {
    asm volatile("s_wait_asynccnt 0x0" ::: "memory");
}
__device__ __forceinline__ unsigned lds_offset(const void* p) {
    // on amdgcn, flat LDS addresses carry the LDS byte offset in addr[31:0]
    return (unsigned)(unsigned long long)p;
}

// ---------------------------------------------------------------------------
// Kernel 1: small linears + accumulator init.
// Flattened regions: [3*BND: msg1,msg2,o1][2*BNT: t1,t2][B*D: msgG+bme]
//                    [B*T: tG+bte][H*16: WtePad][BND: rmax=-FLT_MAX]
// ---------------------------------------------------------------------------
__global__ void prep_kernel(
    const float* __restrict__ node, const float* __restrict__ hidden,
    const float* __restrict__ graph,
    const float* __restrict__ Wm1, const float* __restrict__ bm1,
    const float* __restrict__ Wm2, const float* __restrict__ bm2,
    const float* __restrict__ Wo1, const float* __restrict__ bo1,
    const float* __restrict__ Wt1, const float* __restrict__ bt1,
    const float* __restrict__ Wt2, const float* __restrict__ bt2,
    const float* __restrict__ Wmg, const float* __restrict__ bmg,
    const float* __restrict__ bme,
    const float* __restrict__ Wtg, const float* __restrict__ btg,
    const float* __restrict__ bte,
    const float* __restrict__ Wte,
    float* __restrict__ msg1, float* __restrict__ msg2, float* __restrict__ o1,
    float* __restrict__ t1,  float* __restrict__ t2,
    float* __restrict__ msgG, float* __restrict__ tG,
    float* __restrict__ WtePad, float* __restrict__ rmax)
{
    const int BND = B_ * N_ * D_;
    const int BNT = B_ * N_ * T_;
    long idx = (long)blockIdx.x * blockDim.x + threadIdx.x;

    if (idx < 3L * BND) {                       // msg1 / msg2 / o1  (z @ W + b)
        int r   = (int)(idx / BND);
        int off = (int)(idx % BND);
        int bn  = off / D_, d = off % D_;
        const float* W  = (r == 0) ? Wm1 : (r == 1) ? Wm2 : Wo1;
        const float* bb = (r == 0) ? bm1 : (r == 1) ? bm2 : bo1;
        const float* nf = node   + (long)bn * H_;
        const float* hf = hidden + (long)bn * H_;
        float acc = bb[d];
        for (int k = 0; k < H_; ++k) acc += nf[k] * W[k * D_ + d];
        for (int k = 0; k < H_; ++k) acc += hf[k] * W[(H_ + k) * D_ + d];
        ((r == 0) ? msg1 : (r == 1) ? msg2 : o1)[off] = acc;
        return;
    }
    idx -= 3L * BND;
    if (idx < 2L * BNT) {                       // t1 / t2
        int r   = (int)(idx / BNT);
        int off = (int)(idx % BNT);
        int bn  = off / T_, t = off % T_;
        const float* W  = (r == 0) ? Wt1 : Wt2;
        const float* bb = (r == 0) ? bt1 : bt2;
        const float* nf = node   + (long)bn * H_;
        const float* hf = hidden + (long)bn * H_;
        float acc = bb[t];
        for (int k = 0; k < H_; ++k) acc += nf[k] * W[k * T_ + t];
        for (int k = 0; k < H_; ++k) acc += hf[k] * W[(H_ + k) * T_ + t];
        ((r == 0) ? t1 : t2)[off] = acc;
        return;
    }
    idx -= 2L * BNT;
    if (idx < B_ * D_) {                        // msgG = graph@Wmg + bmg + bme (fold edge bias)
        int b = (int)(idx / D_), d = (int)(idx % D_);
        float acc = bmg[d] + bme[d];
        for (int k = 0; k < H_; ++k) acc += graph[b * H_ + k] * Wmg[k * D_ + d];
        msgG[idx] = acc;
        return;
    }
    idx -= B_ * D_;
    if (idx < B_ * T_) {                        // tG = graph@Wtg + btg + bte
        int b = (int)(idx / T_), t = (int)(idx % T_);
        float acc = btg[t] + bte[t];
        for (int k = 0; k < H_; ++k) acc += graph[b * H_ + k] * Wtg[k * T_ + t];
        tG[idx] = acc;
        return;
    }
    idx -= B_ * T_;
    if (idx < H_ * 16) {                        // Wte zero-padded to 128x16 WMMA B panel
        int k = (int)(idx / 16), t = (int)(idx % 16);
        WtePad[idx] = (t < T_) ? Wte[k * T_ + t] : 0.0f;
        return;
    }
    idx -= H_ * 16;
    if (idx < BND) rmax[idx] = -FLT_MAX;        // max accumulator init
}

// ---------------------------------------------------------------------------
// Kernel 2: fused edge pass with double-buffered async global->LDS staging.
//   msg_e WMMA (waves 0..7) -> masked running-max in registers -> atomic max
//   te    WMMA (wave 8)     -> relu(tri) written directly
// ---------------------------------------------------------------------------
__global__ __launch_bounds__(BDIM) void fused_edge_kernel(
    const float* __restrict__ edge, const float* __restrict__ adj,
    const float* __restrict__ Wme,
    const float* __restrict__ msg1, const float* __restrict__ msg2,
    const float* __restrict__ msgG,
    const float* __restrict__ t1,  const float* __restrict__ t2,
    const float* __restrict__ tG,  const float* __restrict__ WtePad,
    float* __restrict__ rmax, float* __restrict__ out_tri)
{
    __shared__ float eT[2][16 * EPITCH];   // double-buffered 16x128 edge tiles
    __shared__ float adjT[2][16];          // double-buffered adj row slices

    const int tid   = threadIdx.x;
    const int wave  = tid >> 5;
    const int lane  = tid & 31;
    const int col   = lane & 15;     // N within tile
    const int khalf = lane >> 4;     // K/M half select

    const int tiles_j = N_ / 16, chunks = N_ / ICHUNK;
    const int b   = blockIdx.x / (tiles_j * chunks);
    const int rem = blockIdx.x % (tiles_j * chunks);
    const int jbase = (rem / chunks) * 16;
    const int i0    = (rem % chunks) * ICHUNK;

    // --- B panel (128x16) held in registers for the whole i-loop: 32 K-steps ---
    v2f Bk[32];
    if (wave < 8) {
        const float* W = Wme + wave * 16 + col;
        #pragma unroll
        for (int s = 0; s < 32; ++s) {
            int k = 4 * s + 2 * khalf;
            Bk[s].x = W[k * D_];
            Bk[s].y = W[(k + 1) * D_];
        }
    } else {
        const float* W = WtePad + col;
        #pragma unroll
        for (int s = 0; s < 32; ++s) {
            int k = 4 * s + 2 * khalf;
            Bk[s].x = W[k * 16];
            Bk[s].y = W[(k + 1) * 16];
        }
    }

    // --- per-lane invariants in C layout: element r = (M = r + 8*khalf, N = col) ---
    v8f m1t = {}, t1t = {}, racc = {};
    float gv = 0.0f, tgv = 0.0f;
    if (wave < 8) {
        const int dcol = wave * 16 + col;
        gv = msgG[b * D_ + dcol];
        #pragma unroll
        for (int r = 0; r < 8; ++r) {
            int j = jbase + r + 8 * khalf;
            m1t[r]  = msg1[((long)b * N_ + j) * D_ + dcol];
            racc[r] = -FLT_MAX;
        }
    } else if (col < T_) {
        tgv = tG[b * T_ + col];
        #pragma unroll
        for (int r = 0; r < 8; ++r) {
            int j = jbase + r + 8 * khalf;
            t1t[r] = t1[((long)b * N_ + j) * T_ + col];
        }
    }

    const int row16 = tid >> 4;          // loader row (tid < 256)
    const int c8    = (tid & 15) * 8;    // loader col*8 (32B chunk)

    // precomputed loader addresses
    const float* esrc0 = edge + (((long)b * N_ + i0) * N_ + jbase + row16) * H_ + c8;
    const float* asrc0 = adj + ((long)b * N_ + i0) * N_ + jbase + (tid - 256);
    const long   estep = (long)N_ * H_;

    // ---- prologue: async-stage tile i0 into buffer 0 ----
    if (tid < 256) {
        async_ld_b128x2(lds_offset(&eT[0][row16 * EPITCH + c8]), esrc0);
    } else if (tid < 272) {
        async_ld_b32(lds_offset(&adjT[0][tid - 256]), asrc0);
    }

    for (int i = i0; i < i0 + ICHUNK; ++i) {
        const int cur = (i - i0) & 1;

        wait_async0();          // tile i has landed in LDS (this wave's async ops)
        __syncthreads();        // tile i visible to all; buf[1-cur] free to overwrite

        // ---- async-stage tile i+1 into the other buffer (overlaps WMMA below) ----
        if (i + 1 < i0 + ICHUNK) {
            if (tid < 256) {
                async_ld_b128x2(lds_offset(&eT[1 - cur][row16 * EPITCH + c8]),
                                esrc0 + (i + 1 - i0) * estep);
            } else if (tid < 272) {
                async_ld_b32(lds_offset(&adjT[1 - cur][tid - 256]),
                             asrc0 + (i + 1 - i0) * estep);
            }
        }

        // ---- C(16x16) = eT(16x128) x Bpanel(128x16), fp32 WMMA, K-step 4 ----
        v8f c = {};
        const float* arow = &eT[cur][col * EPITCH + 2 * khalf];
        #pragma unroll
        for (int s = 0; s < 32; ++s) {
            v2f a;
            a.x = arow[4 * s];
            a.y = arow[4 * s + 1];
            c = __builtin_amdgcn_wmma_f32_16x16x4_f32(
                    false, a, false, Bk[s], (short)0, c, false, false);
        }

        if (wave < 8) {
            // msgs[b,i,j,d] = msg1[b,j]+msg2[b,i]+msg_e+msgG(+bme), masked, running max over i
            float m2v = msg2[((long)b * N_ + i) * D_ + wave * 16 + col];
            #pragma unroll
            for (int r = 0; r < 8; ++r) {
                float cand = (c[r] + m1t[r] + m2v + gv) * adjT[cur][r + 8 * khalf];
                racc[r] = fmaxf(racc[r], cand);
            }
        } else if (col < T_) {
            // tri[b,i,j,t] = relu(t1[b,j]+t2[b,i]+te+tG(+bte))
            float t2v = t2[((long)b * N_ + i) * T_ + col];
            #pragma unroll
            for (int r = 0; r < 8; ++r) {
                int j = jbase + r + 8 * khalf;
                float v = c[r] + t1t[r] + t2v + tgv;
                out_tri[(((long)b * N_ + i) * N_ + j) * T_ + col] = fmaxf(v, 0.0f);
            }
        }
        // no trailing barrier: next iteration's wait+barrier orders reuse of buf[cur]
    }

    // ---- merge per-chunk maxima across i-chunks ----
    if (wave < 8) {
        const int dcol = wave * 16 + col;
        #pragma unroll
        for (int r = 0; r < 8; ++r) {
            int j = jbase + r + 8 * khalf;
            atomicMaxF32(&rmax[((long)b * N_ + j) * D_ + dcol], racc[r]);
        }
    }
}

// ---------------------------------------------------------------------------
// Kernel 3: ret = o1 + rmax @ Wo2 + bo2   (one block per (b,n) row)
// ---------------------------------------------------------------------------
__global__ __launch_bounds__(D_) void out_kernel(
    const float* __restrict__ rmax, const float* __restrict__ o1,
    const float* __restrict__ Wo2,  const float* __restrict__ bo2,
    float* __restrict__ ret)
{
    __shared__ float rrow[D_];
    const int bn = blockIdx.x;
    const int d  = threadIdx.x;
    rrow[d] = rmax[(long)bn * D_ + d];
    __syncthreads();
    float acc = o1[(long)bn * D_ + d] + bo2[d];
    for (int k = 0; k < D_; ++k) acc += rrow[k] * Wo2[k * D_ + d];
    ret[(long)bn * D_ + d] = acc;
}

// ---------------------------------------------------------------------------
extern "C" void kernel_launch(void* const* d_in, const int* in_sizes, int n_in,
                              void* d_out, int out_size, void* d_ws, size_t ws_size,
                              hipStream_t stream)
{
    const float* node   = (const float*)d_in[0];
    const float* edge   = (const float*)d_in[1];
    const float* graph  = (const float*)d_in[2];
    const float* adj    = (const float*)d_in[3];
    const float* hidden = (const float*)d_in[4];
    const float* Wm1 = (const float*)d_in[5],  *bm1 = (const float*)d_in[6];
    const float* Wm2 = (const float*)d_in[7],  *bm2 = (const float*)d_in[8];
    const float* Wme = (const float*)d_in[9],  *bme = (const float*)d_in[10];
    const float* Wmg = (const float*)d_in[11], *bmg = (const float*)d_in[12];
    const float* Wo1 = (const float*)d_in[13], *bo1 = (const float*)d_in[14];
    const float* Wo2 = (const float*)d_in[15], *bo2 = (const float*)d_in[16];
    const float* Wt1 = (const float*)d_in[17], *bt1 = (const float*)d_in[18];
    const float* Wt2 = (const float*)d_in[19], *bt2 = (const float*)d_in[20];
    const float* Wte = (const float*)d_in[21], *bte = (const float*)d_in[22];
    const float* Wtg = (const float*)d_in[23], *btg = (const float*)d_in[24];

    const int BND = B_ * N_ * D_;
    const int BNT = B_ * N_ * T_;

    float* ws     = (float*)d_ws;
    float* msg1   = ws;  ws += BND;
    float* msg2   = ws;  ws += BND;
    float* o1     = ws;  ws += BND;
    float* t1     = ws;  ws += BNT;
    float* t2     = ws;  ws += BNT;
    float* msgG   = ws;  ws += B_ * D_;
    float* tG     = ws;  ws += B_ * T_;
    float* WtePad = ws;  ws += H_ * 16;
    float* rmaxb  = ws;  ws += BND;

    float* ret = (float*)d_out;          // [B,N,D]
    float* tri = (float*)d_out + BND;    // [B,N,N,T]

    long prep_items = 3L * BND + 2L * BNT + B_ * D_ + B_ * T_ + H_ * 16 + BND;
    int  prep_blocks = (int)((prep_items + 255) / 256);
    prep_kernel<<<prep_blocks, 256, 0, stream>>>(
        node, hidden, graph,
        Wm1, bm1, Wm2, bm2, Wo1, bo1, Wt1, bt1, Wt2, bt2,
        Wmg, bmg, bme, Wtg, btg, bte, Wte,
        msg1, msg2, o1, t1, t2, msgG, tG, WtePad, rmaxb);

    const int grid = B_ * (N_ / 16) * (N_ / ICHUNK);   // 512 workgroups
    fused_edge_kernel<<<grid, BDIM, 0, stream>>>(
        edge, adj, Wme, msg1, msg2, msgG, t1, t2, tG, WtePad, rmaxb, tri);

    out_kernel<<<B_ * N_, D_, 0, stream>>>(rmaxb, o1, Wo2, bo2, ret);
}